// DifferentiableRobotModel_57844619542943
// MI455X (gfx1250) — compile-verified
//
#include <hip/hip_runtime.h>
#include <math.h>

#define N_DOFS 7
#define GRAVF  9.81f

// ---------------------------------------------------------------------------
// Per-joint uniform parameter block (computed once by rnea_prep into d_ws).
// 32 floats = 128 bytes per joint -> read with constant indices in the main
// kernel so the compiler emits SMEM scalar loads (values live in SGPRs).
// ---------------------------------------------------------------------------
struct JParam {
    float rf[9];    // rot_fix, row-major
    float tr[3];    // trans
    float m;        // mass
    float mcom[3];  // mass * com
    float Isp[9];   // inertia + m*(|com|^2 I - com com^T), row-major
    float damping;
    float pad[6];
};

struct F3 { float x, y, z; };

__device__ __forceinline__ F3 f3add(F3 a, F3 b) { return {a.x + b.x, a.y + b.y, a.z + b.z}; }
__device__ __forceinline__ F3 f3sub(F3 a, F3 b) { return {a.x - b.x, a.y - b.y, a.z - b.z}; }
__device__ __forceinline__ F3 f3scale(float s, F3 a) { return {s * a.x, s * a.y, s * a.z}; }
__device__ __forceinline__ F3 f3cross(F3 a, F3 b) {
    return {a.y * b.z - a.z * b.y, a.z * b.x - a.x * b.z, a.x * b.y - a.y * b.x};
}
// R row-major: normal matvec  (R @ v)
__device__ __forceinline__ F3 matv(const float* R, F3 v) {
    return { fmaf(R[0], v.x, fmaf(R[1], v.y, R[2] * v.z)),
             fmaf(R[3], v.x, fmaf(R[4], v.y, R[5] * v.z)),
             fmaf(R[6], v.x, fmaf(R[7], v.y, R[8] * v.z)) };
}
// transposed matvec (R^T @ v)
__device__ __forceinline__ F3 matTv(const float* R, F3 v) {
    return { fmaf(R[0], v.x, fmaf(R[3], v.y, R[6] * v.z)),
             fmaf(R[1], v.x, fmaf(R[4], v.y, R[7] * v.z)),
             fmaf(R[2], v.x, fmaf(R[5], v.y, R[8] * v.z)) };
}
// R = rot_fix @ Rz(c,s):  col0 = c*rf0 + s*rf1, col1 = c*rf1 - s*rf0, col2 = rf2
__device__ __forceinline__ void buildR(float* R, const float* rf, float c, float s) {
    R[0] = fmaf(rf[0], c,  rf[1] * s);
    R[1] = fmaf(rf[1], c, -rf[0] * s);
    R[2] = rf[2];
    R[3] = fmaf(rf[3], c,  rf[4] * s);
    R[4] = fmaf(rf[4], c, -rf[3] * s);
    R[5] = rf[5];
    R[6] = fmaf(rf[6], c,  rf[7] * s);
    R[7] = fmaf(rf[7], c, -rf[6] * s);
    R[8] = rf[8];
}

// ---------------------------------------------------------------------------
// Prep kernel: fold all batch-uniform math (spatial inertia, m*com) out of the
// hot path. Runs once per launch, 7 lanes of one wave32.
// ---------------------------------------------------------------------------
__global__ void rnea_prep(const float* __restrict__ trans,
                          const float* __restrict__ rot_fix,
                          const float* __restrict__ mass,
                          const float* __restrict__ com,
                          const float* __restrict__ inertia,
                          const float* __restrict__ damping,
                          JParam* __restrict__ P) {
    int j = threadIdx.x;
    if (j >= N_DOFS) return;
    JParam p;
#pragma unroll
    for (int r = 0; r < 9; ++r) p.rf[r] = rot_fix[j * 9 + r];
#pragma unroll
    for (int r = 0; r < 3; ++r) p.tr[r] = trans[j * 3 + r];
    float m = mass[j];
    p.m = m;
    float cx = com[j * 3 + 0], cy = com[j * 3 + 1], cz = com[j * 3 + 2];
    p.mcom[0] = m * cx; p.mcom[1] = m * cy; p.mcom[2] = m * cz;
    // Isp = inertia + m * (cx_skew @ cx_skew^T) = inertia + m*(|c|^2 I - c c^T)
    float d = cx * cx + cy * cy + cz * cz;
    p.Isp[0] = inertia[j * 9 + 0] + m * (d - cx * cx);
    p.Isp[1] = inertia[j * 9 + 1] - m * cx * cy;
    p.Isp[2] = inertia[j * 9 + 2] - m * cx * cz;
    p.Isp[3] = inertia[j * 9 + 3] - m * cy * cx;
    p.Isp[4] = inertia[j * 9 + 4] + m * (d - cy * cy);
    p.Isp[5] = inertia[j * 9 + 5] - m * cy * cz;
    p.Isp[6] = inertia[j * 9 + 6] - m * cz * cx;
    p.Isp[7] = inertia[j * 9 + 7] - m * cz * cy;
    p.Isp[8] = inertia[j * 9 + 8] + m * (d - cz * cz);
    p.damping = damping[j];
#pragma unroll
    for (int r = 0; r < 6; ++r) p.pad[r] = 0.0f;
    P[j] = p;
}

// ---------------------------------------------------------------------------
// Main kernel: one thread per batch element, both RNEA sweeps fully unrolled.
// Joint 6 forward state is provably unused (only its rotation matters), so it
// is skipped. Backward sweep rebuilds child rotations from stored (cos,sin).
// Uses hardware v_sin_f32/v_cos_f32 (trans ops co-execute with VALU on CDNA5);
// inputs are N(0,1) so the fast range reduction is numerically safe.
// ---------------------------------------------------------------------------
__global__ __launch_bounds__(256) void rnea_main(const float* __restrict__ q,
                                                 const float* __restrict__ qd,
                                                 const float* __restrict__ qdd,
                                                 const JParam* __restrict__ P,
                                                 float* __restrict__ out,
                                                 int Bn) {
    int b = blockIdx.x * blockDim.x + threadIdx.x;
    if (b >= Bn) return;
    const int base = b * N_DOFS;

    // gfx1250 VMEM prefetch (global_prefetch_b8) for the streamed inputs
    __builtin_prefetch(q + base, 0, 3);
    __builtin_prefetch(qd + base, 0, 3);
    __builtin_prefetch(qdd + base, 0, 3);

    float qv[N_DOFS], qdv[N_DOFS], qddv[N_DOFS];
#pragma unroll
    for (int j = 0; j < N_DOFS; ++j) {
        qv[j]   = q[base + j];
        qdv[j]  = qd[base + j];
        qddv[j] = qdd[base + j];
    }
    float cj[N_DOFS], sj[N_DOFS];
#pragma unroll
    for (int j = 0; j < N_DOFS; ++j) {
        sj[j] = __sinf(qv[j]);   // v_sin_f32 (hardware trans op)
        cj[j] = __cosf(qv[j]);   // v_cos_f32
    }

    // ---------------- forward sweep ----------------
    F3 W[6], Vl[6], DW[6], DV[6];
    F3 wv = {0.f, 0.f, 0.f};
    F3 lv = {0.f, 0.f, 0.f};
    F3 aa = {0.f, 0.f, 0.f};
    F3 la = {0.f, 0.f, GRAVF};
#pragma unroll
    for (int j = 0; j < 6; ++j) {
        const JParam& pj = P[j];
        float R[9];
        buildR(R, pj.rf, cj[j], sj[j]);
        F3 tr = { pj.tr[0], pj.tr[1], pj.tr[2] };
        F3 t_inv = f3scale(-1.0f, matTv(R, tr));
        F3 Rw = matTv(R, wv);
        F3 w  = { Rw.x, Rw.y, Rw.z + qdv[j] };
        F3 v  = f3add(f3cross(t_inv, Rw), matTv(R, lv));
        F3 Ra = matTv(R, aa);
        // dw = Ra + (0,0,qdd) + cross(w, (0,0,qd))
        F3 dw = { fmaf(w.y, qdv[j], Ra.x), fmaf(-w.x, qdv[j], Ra.y), Ra.z + qddv[j] };
        // dv = cross(t_inv, Ra) + R^T la_p + cross(v, (0,0,qd))
        F3 dv = f3add(f3cross(t_inv, Ra), matTv(R, la));
        dv.x = fmaf(v.y, qdv[j], dv.x);
        dv.y = fmaf(-v.x, qdv[j], dv.y);
        W[j] = w; Vl[j] = v; DW[j] = dw; DV[j] = dv;
        wv = w; lv = v; aa = dw; la = dv;
    }

    // ---------------- backward sweep (stores fused in) ----------------
    out[base + 6] = P[6].damping * qdv[6];            // ang_force_z[6] == 0
    F3 lf = {0.f, 0.f, 0.f};
    F3 af = {0.f, 0.f, 0.f};
#pragma unroll
    for (int k = 0; k < 6; ++k) {
        const int j = 5 - k;
        const JParam& pc = P[j + 1];
        float Rc[9];
        buildR(Rc, pc.rf, cj[j + 1], sj[j + 1]);
        F3 Rlf = matv(Rc, lf);
        F3 tc  = { pc.tr[0], pc.tr[1], pc.tr[2] };
        F3 caf = f3add(f3cross(tc, Rlf), matv(Rc, af));   // child angular force
        // body j wrench
        const JParam& pj = P[j];
        float m = pj.m;
        F3 mc = { pj.mcom[0], pj.mcom[1], pj.mcom[2] };
        F3 IcA_l = f3sub(f3scale(m, DV[j]), f3cross(mc, DW[j]));
        F3 IcA_a = f3add(matv(pj.Isp, DW[j]), f3cross(mc, DV[j]));
        F3 IcV_l = f3sub(f3scale(m, Vl[j]), f3cross(mc, W[j]));
        F3 IcV_a = f3add(matv(pj.Isp, W[j]), f3cross(mc, Vl[j]));
        F3 tmp_a = f3add(f3cross(W[j], IcV_a), f3cross(Vl[j], IcV_l));
        F3 tmp_l = f3cross(W[j], IcV_l);
        lf = f3add(f3add(IcA_l, tmp_l), Rlf);
        af = f3add(f3add(IcA_a, tmp_a), caf);
        out[base + j] = fmaf(pj.damping, qdv[j], af.z);
    }
}

// ---------------------------------------------------------------------------
extern "C" void kernel_launch(void* const* d_in, const int* in_sizes, int n_in,
                              void* d_out, int out_size, void* d_ws, size_t ws_size,
                              hipStream_t stream) {
    const float* q       = (const float*)d_in[0];
    const float* qd      = (const float*)d_in[1];
    const float* qdd     = (const float*)d_in[2];
    const float* trans   = (const float*)d_in[3];
    const float* rot_fix = (const float*)d_in[4];
    const float* mass    = (const float*)d_in[5];
    const float* com     = (const float*)d_in[6];
    const float* inertia = (const float*)d_in[7];
    const float* damping = (const float*)d_in[8];
    float* out = (float*)d_out;

    JParam* P = (JParam*)d_ws;   // 7 * 128B = 896B of scratch

    const int Bn = in_sizes[0] / N_DOFS;

    hipLaunchKernelGGL(rnea_prep, dim3(1), dim3(32), 0, stream,
                       trans, rot_fix, mass, com, inertia, damping, P);

    const int block = 256;                       // 8 wave32 per block
    const int grid  = (Bn + block - 1) / block;  // 1024 blocks for B=262144
    hipLaunchKernelGGL(rnea_main, dim3(grid), dim3(block), 0, stream,
                       q, qd, qdd, P, out, Bn);
}